// LinearAndGather_30511447671664
// MI455X (gfx1250) — compile-verified
//
#include <hip/hip_runtime.h>

typedef float v2f __attribute__((ext_vector_type(2)));
typedef float v8f __attribute__((ext_vector_type(8)));

#define D 64

// ---------------------------------------------------------------------------
// WMMA fragment layouts (V_WMMA_F32_16X16X4_F32, wave32):
//  A (16x4):  lane L<16 -> row=L,   holds K=0,1 ; lane L>=16 -> row=L-16, K=2,3
//  B (4x16):  VGPR0: lanes0-15 K=0 col=lane, lanes16-31 K=2 col=lane-16
//             VGPR1: lanes0-15 K=1,          lanes16-31 K=3
//  C/D(16x16):VGPR v, lane L -> row = v + 8*(L>>4), col = L&15
// ---------------------------------------------------------------------------

// Phase 1: Wv^T[m][o] = sum_k W[o,k] * values[k,m], stored so each column m is
// a contiguous 64-float (256 B) record. One wave -> 16 columns (64x16 tile).
// Persistent grid-stride: W fragments loaded once per wave, row offsets hoisted.
__global__ __launch_bounds__(256)
void wv_precompute_kernel(const float* __restrict__ values,
                          const float* __restrict__ weight,
                          float* __restrict__ wv,
                          int Mcols)
{
    const int lane = threadIdx.x & 31;
    const int wave = threadIdx.x >> 5;
    const int row  = lane & 15;   // tile row (A) / tile col (B,D)
    const int kh   = lane >> 4;   // K half select

    // W (16 KB, L2-hot) -> per-wave A fragments, loaded once per wave.
    v2f aw[4][16];
#pragma unroll
    for (int mt = 0; mt < 4; ++mt)
#pragma unroll
        for (int q = 0; q < 16; ++q) {
            const float* wp = weight + (16 * mt + row) * D + 4 * q + 2 * kh;
            v2f a = { wp[0], wp[1] };
            aw[mt][q] = a;
        }

    // Hoist the 16 per-K-chunk row offsets (loop-invariant across tiles):
    // removes all per-tile 64-bit multiplies from the hot loop.
    size_t rowoff[16];
#pragma unroll
    for (int q = 0; q < 16; ++q)
        rowoff[q] = (size_t)(4 * q + 2 * kh) * (size_t)Mcols;

    const int tiles  = (Mcols + 15) >> 4;
    const int stride = gridDim.x * 8;

    for (int t = blockIdx.x * 8 + wave; t < tiles; t += stride) {
        const int m0 = t * 16;
        int mc = m0 + row; if (mc >= Mcols) mc = Mcols - 1;   // clamp tail loads
        const float* colbase = values + mc;

        v8f acc[4] = {};
#pragma unroll
        for (int q = 0; q < 16; ++q) {
            const float* vp = colbase + rowoff[q];
            v2f b = { vp[0], vp[Mcols] };   // coalesced: contiguous in m
#pragma unroll
            for (int mt = 0; mt < 4; ++mt)
                acc[mt] = __builtin_amdgcn_wmma_f32_16x16x4_f32(
                    false, aw[mt][q], false, b, (short)0, acc[mt], false, false);
        }

        if (m0 + row < Mcols) {
            // Lane writes 8 consecutive floats per tile -> 32 B vector stores.
            float* op = wv + (size_t)(m0 + row) * D + 8 * kh;
#pragma unroll
            for (int mt = 0; mt < 4; ++mt)
                *(v8f*)(op + 16 * mt) = acc[mt];
        }
    }
}

// Phase 2: out[o, j] = Wv^T[in_idx[out_idx[j]]][o]; contiguous 256 B gathered
// reads per thread (L2-friendly: 2e6 gathers over 1e6 records, 192 MB L2),
// fully lane-coalesced strided writes.
__global__ __launch_bounds__(256)
void gather_kernel(const float* __restrict__ wv,
                   const int* __restrict__ in_idx,
                   const int* __restrict__ out_idx,
                   float* __restrict__ out,
                   int n_out)
{
    const int j = blockIdx.x * 256 + threadIdx.x;
    if (j >= n_out) return;
    const int c = in_idx[out_idx[j]];
    const v8f* src = (const v8f*)(wv + (size_t)c * D);
#pragma unroll
    for (int g = 0; g < 8; ++g) {
        v8f v = src[g];
#pragma unroll
        for (int e = 0; e < 8; ++e)
            out[(size_t)(8 * g + e) * (size_t)n_out + j] = v[e];
    }
}

// Fallback (scratch too small): fused gather + WMMA GEMM per 64x16 out tile.
__global__ __launch_bounds__(256)
void fused_kernel(const float* __restrict__ values,
                  const float* __restrict__ weight,
                  const int* __restrict__ in_idx,
                  const int* __restrict__ out_idx,
                  float* __restrict__ out,
                  int Mcols, int n_out)
{
    const int lane = threadIdx.x & 31;
    const int wave = threadIdx.x >> 5;
    const int row  = lane & 15;
    const int kh   = lane >> 4;

    v2f aw[4][16];
#pragma unroll
    for (int mt = 0; mt < 4; ++mt)
#pragma unroll
        for (int q = 0; q < 16; ++q) {
            const float* wp = weight + (16 * mt + row) * D + 4 * q + 2 * kh;
            v2f a = { wp[0], wp[1] };
            aw[mt][q] = a;
        }

    size_t rowoff[16];
#pragma unroll
    for (int q = 0; q < 16; ++q)
        rowoff[q] = (size_t)(4 * q + 2 * kh) * (size_t)Mcols;

    const int tiles  = (n_out + 15) >> 4;
    const int stride = gridDim.x * 8;

    for (int t = blockIdx.x * 8 + wave; t < tiles; t += stride) {
        const int j0 = t * 16;
        int jc = j0 + row; if (jc >= n_out) jc = n_out - 1;
        const int c = in_idx[out_idx[jc]];      // gathered `values` column
        const float* colbase = values + (size_t)c;

        v8f acc[4] = {};
#pragma unroll
        for (int q = 0; q < 16; ++q) {
            const float* vp = colbase + rowoff[q];
            v2f b = { vp[0], vp[Mcols] };
#pragma unroll
            for (int mt = 0; mt < 4; ++mt)
                acc[mt] = __builtin_amdgcn_wmma_f32_16x16x4_f32(
                    false, aw[mt][q], false, b, (short)0, acc[mt], false, false);
        }

        if (j0 + row < n_out) {
#pragma unroll
            for (int mt = 0; mt < 4; ++mt)
#pragma unroll
                for (int v = 0; v < 8; ++v)
                    out[(size_t)(16 * mt + 8 * kh + v) * (size_t)n_out + (j0 + row)]
                        = acc[mt][v];
        }
    }
}

extern "C" void kernel_launch(void* const* d_in, const int* in_sizes, int n_in,
                              void* d_out, int out_size, void* d_ws, size_t ws_size,
                              hipStream_t stream) {
    const float* values  = (const float*)d_in[0];   // (64, M) row-major
    const float* weight  = (const float*)d_in[1];   // (64, 64) row-major
    const int*   in_idx  = (const int*)d_in[2];     // (N_IN,)
    const int*   out_idx = (const int*)d_in[3];     // (N_OUT,)
    float*       out     = (float*)d_out;           // (64, N_OUT) row-major

    const int Mcols = in_sizes[0] / D;
    const int n_out = in_sizes[3];

    const size_t need = (size_t)Mcols * D * sizeof(float);
    if (ws_size >= need) {
        const int tiles  = (Mcols + 15) / 16;
        int blocks = (tiles + 7) / 8;
        if (blocks > 640) blocks = 640;               // persistent waves
        wv_precompute_kernel<<<blocks, 256, 0, stream>>>(
            values, weight, (float*)d_ws, Mcols);
        const int gblocks = (n_out + 255) / 256;
        gather_kernel<<<gblocks, 256, 0, stream>>>(
            (const float*)d_ws, in_idx, out_idx, out, n_out);
    } else {
        const int tiles  = (n_out + 15) / 16;
        int blocks = (tiles + 7) / 8;
        if (blocks > 640) blocks = 640;
        fused_kernel<<<blocks, 256, 0, stream>>>(
            values, weight, in_idx, out_idx, out, Mcols, n_out);
    }
}